// DisentangledSelfAttention_47261820125678
// MI455X (gfx1250) — compile-verified
//
#include <hip/hip_runtime.h>
#include <hip/hip_bf16.h>
#include <cmath>

// Problem constants (DeBERTa disentangled attention reference)
#define BATCH 2
#define SEQ   1024
#define HEADS 16
#define HDIM  64
#define HID   1024          // HEADS*HDIM
#define K2    512           // 2*K buckets
#define NEG_BIG (-3.4028234663852886e38f)

typedef float v2f __attribute__((ext_vector_type(2)));
typedef float v8f __attribute__((ext_vector_type(8)));

// ---------------------------------------------------------------------------
// Generic fp32 WMMA GEMM:  C = scale * (A @ op(B)) + bias
//   BT=true : B is (N x Kc) row-major  -> C = A @ B^T   (NT)
//   BT=false: B is (Kc x N) row-major  -> C = A @ B     (NN)
// Workgroup: 256 threads (8 wave32), output tile 128(M) x 64(N).
// Each wave owns a 16x64 slab -> 4 accumulators of v8f.
// Batched via grid.z with outer/inner offset decomposition (z = outer*innerCnt+inner).
// ---------------------------------------------------------------------------
template <bool BT>
__global__ __launch_bounds__(256) void gemm_wmma_f32(
    const float* __restrict__ A, int lda,
    const float* __restrict__ Bm, int ldb,
    const float* __restrict__ bias,
    float* __restrict__ C, int ldc,
    int M, int N, int Kc, float scale,
    long aOuter, long aInner, long bOuter, long bInner,
    long cOuter, long cInner, int innerCnt)
{
    const int tid  = threadIdx.x;
    const int lane = tid & 31;
    const int wave = tid >> 5;
    const int llo  = lane & 15;
    const int lhi  = lane >> 4;

    const int z  = blockIdx.z;
    const int zo = z / innerCnt;
    const int zi = z % innerCnt;
    const float* Ab = A  + (size_t)zo * aOuter + (size_t)zi * aInner;
    const float* Bb = Bm + (size_t)zo * bOuter + (size_t)zi * bInner;
    float*       Cb = C  + (size_t)zo * cOuter + (size_t)zi * cInner;

    const int m0 = blockIdx.y * 128;
    const int n0 = blockIdx.x * 64;

    __shared__ float As[128 * 34];   // [r][k], stride 34 (8B-aligned rows, bank-spread)
    __shared__ float Bs[64 * 34];    // BT: [n][k] stride 34 ; NN: [k][n] stride 66 (32*66=2112<=2176)

    v8f acc[4] = {};

    for (int k0 = 0; k0 < Kc; k0 += 32) {
        // Stage A tile (128 x 32)
        for (int idx = tid; idx < 128 * 32; idx += 256) {
            int r = idx >> 5, c = idx & 31;
            As[r * 34 + c] = Ab[(size_t)(m0 + r) * lda + (k0 + c)];
        }
        // Stage B tile
        if (BT) {
            for (int idx = tid; idx < 64 * 32; idx += 256) {
                int r = idx >> 5, c = idx & 31;          // r = n, c = k
                Bs[r * 34 + c] = Bb[(size_t)(n0 + r) * ldb + (k0 + c)];
            }
        } else {
            for (int idx = tid; idx < 32 * 64; idx += 256) {
                int r = idx >> 6, c = idx & 63;          // r = k, c = n
                Bs[r * 66 + c] = Bb[(size_t)(k0 + r) * ldb + (n0 + c)];
            }
        }
        __syncthreads();

        const int arow = (wave * 16 + llo) * 34;
#pragma unroll
        for (int kk = 0; kk < 32; kk += 4) {
            v2f a;
            a[0] = As[arow + kk + 2 * lhi];
            a[1] = As[arow + kk + 2 * lhi + 1];
#pragma unroll
            for (int nt = 0; nt < 4; ++nt) {
                v2f b;
                if (BT) {
                    int boff = (nt * 16 + llo) * 34 + kk + 2 * lhi;
                    b[0] = Bs[boff];
                    b[1] = Bs[boff + 1];
                } else {
                    int boff = (kk + 2 * lhi) * 66 + nt * 16 + llo;
                    b[0] = Bs[boff];
                    b[1] = Bs[boff + 66];
                }
                acc[nt] = __builtin_amdgcn_wmma_f32_16x16x4_f32(
                    false, a, false, b, (short)0, acc[nt], false, false);
            }
        }
        __syncthreads();
    }

    // Epilogue: D layout — element (v,lane): M = v + 8*lhi, N = llo
#pragma unroll
    for (int nt = 0; nt < 4; ++nt) {
        int col = n0 + nt * 16 + llo;
        float bi = bias ? bias[col] : 0.0f;
#pragma unroll
        for (int vv = 0; vv < 8; ++vv) {
            int row = m0 + wave * 16 + vv + 8 * lhi;
            if (row < M && col < N)
                Cb[(size_t)row * ldc + col] = acc[nt][vv] * scale + bi;
        }
    }
}

// ---------------------------------------------------------------------------
// Fused scores + disentangled-bias gather:
//   logits[b,h,i,j] = scale*(q_i . k_j) + c2p[b,h,i, clip(rp[i,j]+256)]
//                                       + p2c[b,h,j, clip(-rp[j,i]+256)]
// (c2p/p2c are pre-scaled). 128x64 tile per workgroup, Kc = HDIM = 64.
// ---------------------------------------------------------------------------
__global__ __launch_bounds__(256) void logits_wmma(
    const float* __restrict__ q, const float* __restrict__ k,
    const float* __restrict__ c2p, const float* __restrict__ p2c,
    const int* __restrict__ rp, float* __restrict__ logits, float scale)
{
    const int tid  = threadIdx.x;
    const int lane = tid & 31;
    const int wave = tid >> 5;
    const int llo  = lane & 15;
    const int lhi  = lane >> 4;

    const int z = blockIdx.z;            // z = b*HEADS + h
    const int b = z / HEADS;
    const int h = z % HEADS;
    const float* Ab = q + (size_t)b * SEQ * HID + h * HDIM;
    const float* Bb = k + (size_t)b * SEQ * HID + h * HDIM;

    const int m0 = blockIdx.y * 128;
    const int n0 = blockIdx.x * 64;

    __shared__ float As[128 * 66];       // q rows (i) x 64
    __shared__ float Bs[64 * 66];        // k rows (j) x 64

    for (int idx = tid; idx < 128 * 64; idx += 256) {
        int r = idx >> 6, c = idx & 63;
        As[r * 66 + c] = Ab[(size_t)(m0 + r) * HID + c];
    }
    for (int idx = tid; idx < 64 * 64; idx += 256) {
        int r = idx >> 6, c = idx & 63;
        Bs[r * 66 + c] = Bb[(size_t)(n0 + r) * HID + c];
    }
    __syncthreads();

    v8f acc[4] = {};
    const int arow = (wave * 16 + llo) * 66;
#pragma unroll
    for (int kk = 0; kk < 64; kk += 4) {
        v2f a;
        a[0] = As[arow + kk + 2 * lhi];
        a[1] = As[arow + kk + 2 * lhi + 1];
#pragma unroll
        for (int nt = 0; nt < 4; ++nt) {
            int boff = (nt * 16 + llo) * 66 + kk + 2 * lhi;
            v2f bfr;
            bfr[0] = Bs[boff];
            bfr[1] = Bs[boff + 1];
            acc[nt] = __builtin_amdgcn_wmma_f32_16x16x4_f32(
                false, a, false, bfr, (short)0, acc[nt], false, false);
        }
    }

    const size_t zbase = (size_t)z * SEQ;
#pragma unroll
    for (int nt = 0; nt < 4; ++nt) {
        int j = n0 + nt * 16 + llo;
#pragma unroll
        for (int vv = 0; vv < 8; ++vv) {
            int i = m0 + wave * 16 + vv + 8 * lhi;
            int r_ij = rp[(size_t)i * SEQ + j];
            int r_ji = rp[(size_t)j * SEQ + i];
            int ci = min(max(r_ij + 256, 0), K2 - 1);
            int pi = min(max(-r_ji + 256, 0), K2 - 1);
            float val = acc[nt][vv] * scale
                      + c2p[(zbase + i) * K2 + ci]
                      + p2c[(zbase + j) * K2 + pi];
            logits[(zbase + i) * SEQ + j] = val;
        }
    }
}

// ---------------------------------------------------------------------------
// Row softmax with reference semantics:
//   logits_out = raw - max(raw)          (unmasked max, written in place)
//   probs      = where(mask, softmax(where(mask, logits_out, -BIG)), 0)
// One 256-thread block per row (B*H*S rows, row length SEQ=1024).
// ---------------------------------------------------------------------------
__device__ __forceinline__ float blk_reduce_max(float v, float* sred, int tid) {
    sred[tid] = v; __syncthreads();
    for (int s = 128; s > 0; s >>= 1) {
        if (tid < s) sred[tid] = fmaxf(sred[tid], sred[tid + s]);
        __syncthreads();
    }
    float r = sred[0]; __syncthreads();
    return r;
}
__device__ __forceinline__ float blk_reduce_sum(float v, float* sred, int tid) {
    sred[tid] = v; __syncthreads();
    for (int s = 128; s > 0; s >>= 1) {
        if (tid < s) sred[tid] += sred[tid + s];
        __syncthreads();
    }
    float r = sred[0]; __syncthreads();
    return r;
}

__global__ __launch_bounds__(256) void softmax_rows(
    float* __restrict__ logits, const int* __restrict__ mask,
    float* __restrict__ probs)
{
    const int row = blockIdx.x;                 // (b*HEADS + h)*SEQ + i
    const int tid = threadIdx.x;
    const int b   = row / (HEADS * SEQ);
    const int i   = row % SEQ;
    const size_t lbase = (size_t)row * SEQ;
    const size_t mbase = (size_t)b * SEQ * SEQ + (size_t)i * SEQ;

    __shared__ float sred[256];

    float lv[4]; int mv[4];
    float mx = NEG_BIG;
#pragma unroll
    for (int u = 0; u < 4; ++u) {
        int j = tid + u * 256;
        lv[u] = logits[lbase + j];
        mv[u] = mask[mbase + j];
        mx = fmaxf(mx, lv[u]);
    }
    const float rowmax = blk_reduce_max(mx, sred, tid);   // unmasked max

    float mmx = NEG_BIG;
#pragma unroll
    for (int u = 0; u < 4; ++u) {
        lv[u] -= rowmax;
        logits[lbase + tid + u * 256] = lv[u];            // shifted logits out
        if (mv[u]) mmx = fmaxf(mmx, lv[u]);
    }
    const float m2 = blk_reduce_max(mmx, sred, tid);      // masked max

    float ls = 0.0f;
    float ev[4];
#pragma unroll
    for (int u = 0; u < 4; ++u) {
        ev[u] = mv[u] ? __expf(lv[u] - m2) : 0.0f;
        ls += ev[u];
    }
    const float denom = blk_reduce_sum(ls, sred, tid);
    const float inv = 1.0f / denom;
#pragma unroll
    for (int u = 0; u < 4; ++u)
        probs[lbase + tid + u * 256] = ev[u] * inv;
}

// ---------------------------------------------------------------------------
// Host-side launch
// ---------------------------------------------------------------------------
extern "C" void kernel_launch(void* const* d_in, const int* in_sizes, int n_in,
                              void* d_out, int out_size, void* d_ws, size_t ws_size,
                              hipStream_t stream) {
    (void)in_sizes; (void)n_in; (void)out_size; (void)ws_size;

    const float* hs   = (const float*)d_in[0];   // (B,S,HID)
    const int*   mask = (const int*)  d_in[1];   // (B,1,S,S)
    const int*   rp   = (const int*)  d_in[2];   // (S,S)
    const float* rel  = (const float*)d_in[3];   // (2K,HID)
    const float* Wq   = (const float*)d_in[4];
    const float* bq   = (const float*)d_in[5];
    const float* Wk   = (const float*)d_in[6];
    const float* bk   = (const float*)d_in[7];
    const float* Wv   = (const float*)d_in[8];
    const float* bv   = (const float*)d_in[9];

    const size_t QKV = (size_t)BATCH * SEQ * HID;      // 2,097,152
    const size_t POS = (size_t)K2 * HID;               //   524,288
    const size_t CP  = (size_t)BATCH * HEADS * SEQ * K2; // 16,777,216

    float* q   = (float*)d_ws;
    float* kbuf= q   + QKV;
    float* v   = kbuf+ QKV;
    float* PK  = v   + QKV;
    float* PQ  = PK  + POS;
    float* c2p = PQ  + POS;
    float* p2c = c2p + CP;

    float* ctx    = (float*)d_out;                              // (B,S,HID)
    float* probs  = ctx   + QKV;                                // (B,H,S,S)
    float* logits = probs + (size_t)BATCH * HEADS * SEQ * SEQ;  // (B,H,S,S)

    const float scale = 1.0f / sqrtf((float)(HDIM * 3));

    dim3 blk(256);

    // 1) Q/K/V projections: (2048 x 1024) = hs @ W^T + b
    {
        dim3 g(HID / 64, (BATCH * SEQ) / 128, 1);
        gemm_wmma_f32<true><<<g, blk, 0, stream>>>(hs, HID, Wq, HID, bq, q,    HID,
            BATCH * SEQ, HID, HID, 1.0f, 0, 0, 0, 0, 0, 0, 1);
        gemm_wmma_f32<true><<<g, blk, 0, stream>>>(hs, HID, Wk, HID, bk, kbuf, HID,
            BATCH * SEQ, HID, HID, 1.0f, 0, 0, 0, 0, 0, 0, 1);
        gemm_wmma_f32<true><<<g, blk, 0, stream>>>(hs, HID, Wv, HID, bv, v,    HID,
            BATCH * SEQ, HID, HID, 1.0f, 0, 0, 0, 0, 0, 0, 1);
    }
    // 2) Positional projections: (512 x 1024) = rel @ W^T + b
    {
        dim3 g(HID / 64, K2 / 128, 1);
        gemm_wmma_f32<true><<<g, blk, 0, stream>>>(rel, HID, Wk, HID, bk, PK, HID,
            K2, HID, HID, 1.0f, 0, 0, 0, 0, 0, 0, 1);
        gemm_wmma_f32<true><<<g, blk, 0, stream>>>(rel, HID, Wq, HID, bq, PQ, HID,
            K2, HID, HID, 1.0f, 0, 0, 0, 0, 0, 0, 1);
    }
    // 3) c2p = scale * q_head @ PK_head^T ; p2c = scale * k_head @ PQ_head^T
    //    per (b,h): M=SEQ, N=K2, Kc=HDIM. batch 32 = outer(b) x inner(h).
    {
        dim3 g(K2 / 64, SEQ / 128, BATCH * HEADS);
        long aO = (long)SEQ * HID, aI = HDIM;
        long bO = 0,               bI = HDIM;
        long cO = (long)HEADS * SEQ * K2, cI = (long)SEQ * K2;
        gemm_wmma_f32<true><<<g, blk, 0, stream>>>(q,    HID, PK, HID, nullptr, c2p, K2,
            SEQ, K2, HDIM, scale, aO, aI, bO, bI, cO, cI, HEADS);
        gemm_wmma_f32<true><<<g, blk, 0, stream>>>(kbuf, HID, PQ, HID, nullptr, p2c, K2,
            SEQ, K2, HDIM, scale, aO, aI, bO, bI, cO, cI, HEADS);
    }
    // 4) Raw logits = scale*(q.k^T) + gathered c2p + gathered p2c
    {
        dim3 g(SEQ / 64, SEQ / 128, BATCH * HEADS);
        logits_wmma<<<g, blk, 0, stream>>>(q, kbuf, c2p, p2c, rp, logits, scale);
    }
    // 5) Row-wise max subtraction (in place on logits) + masked softmax -> probs
    {
        dim3 g(BATCH * HEADS * SEQ);
        softmax_rows<<<g, blk, 0, stream>>>(logits, mask, probs);
    }
    // 6) ctx = probs @ v_head (NN): per (b,h): M=SEQ, N=HDIM, Kc=SEQ
    {
        dim3 g(1, SEQ / 128, BATCH * HEADS);
        long aO = (long)HEADS * SEQ * SEQ, aI = (long)SEQ * SEQ;
        long bO = (long)SEQ * HID,         bI = HDIM;
        long cO = (long)SEQ * HID,         cI = HDIM;
        gemm_wmma_f32<false><<<g, blk, 0, stream>>>(probs, SEQ, v, HID, nullptr, ctx, HID,
            SEQ, HDIM, SEQ, 1.0f, aO, aI, bO, bI, cO, cI, HEADS);
    }
}